// BindingSiteGNN_90494960927382
// MI455X (gfx1250) — compile-verified
//
#include <hip/hip_runtime.h>
#include <hip/hip_bf16.h>
#include <cstdint>
#include <cstddef>

// ---------------------------------------------------------------------------
// BindingSiteGNN forward on gfx1250 (CDNA5).
// Dense layers: V_WMMA_F32_16X16X4_F32 (full fp32 matrix cores), K-pair
// interleaved B tiles in LDS so every B fragment is one ds_load_b64.
// Edge phase:   float4-vectorized gather/scatter with f32 atomics.
// ---------------------------------------------------------------------------

typedef __attribute__((ext_vector_type(2))) float v2f;
typedef __attribute__((ext_vector_type(8))) float v8f;

#define BN_RS 0.9999950000374997f   // rsqrt(1 + 1e-5)

__device__ __forceinline__ float elu1(float v) { return v > 0.f ? v : (__expf(v) - 1.f); }

__device__ __forceinline__ unsigned f2mono(float f) {
    unsigned u = __float_as_uint(f);
    return (u & 0x80000000u) ? ~u : (u | 0x80000000u);
}
__device__ __forceinline__ float mono2f(unsigned u) {
    unsigned b = (u & 0x80000000u) ? (u ^ 0x80000000u) : ~u;
    return __uint_as_float(b);
}

// ---------------------------------------------------------------------------
// WMMA fp32 GEMM: D[M,Nout] = epilogue(A[M,K] @ B[K,Nout])
// 256 threads = 8 waves. Wave w -> rows [blk*128 + w*16, +16), cols [bly*64, +64)
// (4 accumulators of 16x16).
// LDS B tile: K-chunk of 32 stored as 16 k-pairs x 64 cols of float2
//   BsP[kp*80 + n] = (B[k0+2kp][nBase+n], B[k0+2kp+1][nBase+n])
// pair-row stride 80 -> 160 dwords -> lanes 16-31 shifted 32 banks: conflict-free.
// ACT: 0=none 1=relu 2=elu.  BN: v = v*g*RS + bb.  RES: v += res.  Order: bias,BN,RES,ACT.
// ---------------------------------------------------------------------------
#define SPAIR 80   // float2 stride between k-pair rows

template <int ACT, bool BN, bool RES, bool BIAS>
__global__ __launch_bounds__(256) void gemm_wmma_f32(
    const float* __restrict__ A, const float* __restrict__ B,
    const float* __restrict__ bias, const float* __restrict__ g,
    const float* __restrict__ bb, const float* __restrict__ res,
    float* __restrict__ D, int M, int K, int Nout)
{
    __shared__ float BsF[16 * SPAIR * 2];   // 16 k-pairs x 80 float2 = 10240 B

    const int tid  = threadIdx.x;
    const int wave = tid >> 5;
    const int lane = tid & 31;
    const int l16  = lane & 15;
    const int lp   = lane >> 4;                 // 0 or 1 (K parity half)
    const int mBase = blockIdx.x * 128 + wave * 16;
    const int nBase = blockIdx.y * 64;

    v8f acc[4];
#pragma unroll
    for (int j = 0; j < 4; ++j)
#pragma unroll
        for (int r = 0; r < 8; ++r) acc[j][r] = 0.f;

    const int rowA = mBase + l16;
    const int rAcl = rowA < M ? rowA : (M - 1);   // clamp: discarded on store guard
    const float* Arow = A + (size_t)rAcl * K;

    // staging coords: each thread owns (k-pair, 4-col group)
    const int skp = tid >> 4;          // 0..15
    const int scg = (tid & 15) * 4;    // 0..60 step 4

    for (int k0 = 0; k0 < K; k0 += 32) {
        __syncthreads();
        {
            const float* b0 = B + (size_t)(k0 + 2 * skp)     * Nout + nBase + scg;
            const float* b1 = B + (size_t)(k0 + 2 * skp + 1) * Nout + nBase + scg;
            float4 u = *(const float4*)b0;     // global_load_b128
            float4 v = *(const float4*)b1;     // global_load_b128
            float* dst = BsF + (skp * SPAIR + scg) * 2;
            float4 p0, p1;
            p0.x = u.x; p0.y = v.x; p0.z = u.y; p0.w = v.y;
            p1.x = u.z; p1.y = v.z; p1.z = u.w; p1.w = v.w;
            *(float4*)(dst + 0) = p0;          // ds_store_b128
            *(float4*)(dst + 4) = p1;          // ds_store_b128
        }
        __syncthreads();
#pragma unroll
        for (int kk = 0; kk < 32; kk += 4) {
            const int ka = k0 + kk + 2 * lp;
            v2f a = *(const v2f*)(Arow + ka);    // global_load_b64
            const float* bbase = BsF + (((kk >> 1) + lp) * SPAIR + l16) * 2;
#pragma unroll
            for (int j = 0; j < 4; ++j) {
                v2f b = *(const v2f*)(bbase + j * 32);   // ds_load_b64
                acc[j] = __builtin_amdgcn_wmma_f32_16x16x4_f32(
                    false, a, false, b, (short)0, acc[j], false, false);
            }
        }
    }

    const int rowBase = mBase + lp * 8;
#pragma unroll
    for (int j = 0; j < 4; ++j) {
        const int col = nBase + j * 16 + l16;
        float bi = 0.f, gg = 1.f, bv = 0.f;
        if constexpr (BIAS) bi = bias[col];
        if constexpr (BN) { gg = g[col] * BN_RS; bv = bb[col]; }
#pragma unroll
        for (int r = 0; r < 8; ++r) {
            const int row = rowBase + r;
            if (row < M) {
                float v = acc[j][r] + bi;
                if constexpr (BN)  v = v * gg + bv;
                if constexpr (RES) v += res[(size_t)row * Nout + col];
                if constexpr (ACT == 1) v = v > 0.f ? v : 0.f;
                if constexpr (ACT == 2) v = elu1(v);
                D[(size_t)row * Nout + col] = v;
            }
        }
    }
}

// ---------------------------------------------------------------------------
// Edge-phase kernels
// ---------------------------------------------------------------------------
__global__ void deg_kernel(const int* __restrict__ ei, float* deg, float* cnt, int E, int N)
{
    int i = blockIdx.x * blockDim.x + threadIdx.x;
    if (i >= E + N) return;
    int d = (i < E) ? ei[E + i] : (i - E);
    atomicAdd(&deg[d], 1.f);
    if (i < E) atomicAdd(&cnt[d], 1.f);
}

__global__ void dinv_kernel(const float* __restrict__ deg, const float* __restrict__ cnt,
                            float* dinv, float* cinv, int N)
{
    int i = blockIdx.x * blockDim.x + threadIdx.x;
    if (i >= N) return;
    dinv[i] = rsqrtf(fmaxf(deg[i], 1.f));
    cinv[i] = 1.f / fmaxf(cnt[i], 1.f);
}

// a_src[n,h] = sum_c hg[n,h*128+c]*att_src[h*128+c] ; one wave per node
__global__ __launch_bounds__(256) void att_kernel(
    const float* __restrict__ hg, const float* __restrict__ att_s,
    const float* __restrict__ att_d, float* asrc, float* adst, int N)
{
    int wid  = (blockIdx.x * blockDim.x + threadIdx.x) >> 5;
    int lane = threadIdx.x & 31;
    if (wid >= N) return;
    const float* row = hg + (size_t)wid * 512;
#pragma unroll
    for (int hd = 0; hd < 4; ++hd) {
        float ss = 0.f, sd = 0.f;
#pragma unroll
        for (int c = lane; c < 128; c += 32) {
            float v = row[hd * 128 + c];
            ss += v * att_s[hd * 128 + c];
            sd += v * att_d[hd * 128 + c];
        }
#pragma unroll
        for (int m = 16; m > 0; m >>= 1) {
            ss += __shfl_xor(ss, m, 32);
            sd += __shfl_xor(sd, m, 32);
        }
        if (lane == 0) { asrc[wid * 4 + hd] = ss; adst[wid * 4 + hd] = sd; }
    }
}

// pass1: e = leaky_relu(a_src[s]+a_dst[d]); segment max via monotone-uint atomicMax
__global__ void gat_logits_max(const int* __restrict__ ei, const float* __restrict__ asrc,
                               const float* __restrict__ adst, float* ebuf,
                               unsigned* emax, int E, int N)
{
    int i = blockIdx.x * blockDim.x + threadIdx.x;
    if (i >= (E + N) * 4) return;
    int e = i >> 2, hd = i & 3;
    int s = (e < E) ? ei[e]     : (e - E);
    int d = (e < E) ? ei[E + e] : (e - E);
    float v = asrc[s * 4 + hd] + adst[d * 4 + hd];
    v = v > 0.f ? v : 0.2f * v;
    ebuf[i] = v;
    atomicMax(&emax[d * 4 + hd], f2mono(v));
}

// pass2: ex = exp(e - emax[d]); denom[d] += ex
__global__ void gat_expsum(const int* __restrict__ ei, float* ebuf,
                           const unsigned* __restrict__ emax, float* denom, int E, int N)
{
    int i = blockIdx.x * blockDim.x + threadIdx.x;
    if (i >= (E + N) * 4) return;
    int e = i >> 2, hd = i & 3;
    int d = (e < E) ? ei[E + e] : (e - E);
    float ex = __expf(ebuf[i] - mono2f(emax[d * 4 + hd]));
    ebuf[i] = ex;
    atomicAdd(&denom[d * 4 + hd], ex);
}

// pass3: alpha = ex/(denom+1e-16); also GCN edge norm (one thread per edge)
__global__ void gat_alpha(const int* __restrict__ ei, float* ebuf,
                          const float* __restrict__ denom,
                          const float* __restrict__ dinv, float* normE, int E, int N)
{
    int i = blockIdx.x * blockDim.x + threadIdx.x;
    if (i >= (E + N) * 4) return;
    int e = i >> 2, hd = i & 3;
    int s = (e < E) ? ei[e]     : (e - E);
    int d = (e < E) ? ei[E + e] : (e - E);
    ebuf[i] = ebuf[i] / (denom[d * 4 + hd] + 1e-16f);
    if (hd == 0) normE[e] = dinv[s] * dinv[d];
}

// fused scatter: GAT messages, GCN normalized sum, SAGE sum.
// 4 channels per thread (float4 gathers, 128 threads/edge).
__global__ void scatter_kernel(const int* __restrict__ ei,
                               const float* __restrict__ hg, const float* __restrict__ xw,
                               const float* __restrict__ h,  const float* __restrict__ alpha,
                               const float* __restrict__ normE,
                               float* x1, float* x2, float* agg, int E, int N)
{
    long long i = (long long)blockIdx.x * blockDim.x + threadIdx.x;
    long long tot = (long long)(E + N) * 128;
    if (i >= tot) return;
    int e = (int)(i >> 7);
    int c = (int)(i & 127) * 4;
    int s = (e < E) ? ei[e]     : (e - E);
    int d = (e < E) ? ei[E + e] : (e - E);
    float  al = alpha[e * 4 + (c >> 7)];
    float  nr = normE[e];
    float4 vg = *(const float4*)(hg + (size_t)s * 512 + c);
    float4 vx = *(const float4*)(xw + (size_t)s * 512 + c);
    float* p1 = x1 + (size_t)d * 512 + c;
    float* p2 = x2 + (size_t)d * 512 + c;
    atomicAdd(p1 + 0, vg.x * al); atomicAdd(p1 + 1, vg.y * al);
    atomicAdd(p1 + 2, vg.z * al); atomicAdd(p1 + 3, vg.w * al);
    atomicAdd(p2 + 0, vx.x * nr); atomicAdd(p2 + 1, vx.y * nr);
    atomicAdd(p2 + 2, vx.z * nr); atomicAdd(p2 + 3, vx.w * nr);
    if (e < E) {
        float4 vh = *(const float4*)(h + (size_t)s * 512 + c);
        float* pa = agg + (size_t)d * 512 + c;
        atomicAdd(pa + 0, vh.x); atomicAdd(pa + 1, vh.y);
        atomicAdd(pa + 2, vh.z); atomicAdd(pa + 3, vh.w);
    }
}

// x1 = elu(x1+b_gat); x2 = elu(x2+b_gcn); agg *= 1/max(cnt,1)   (float4)
__global__ void post_scatter(float* x1, float* x2, float* agg,
                             const float* __restrict__ bgat, const float* __restrict__ bgcn,
                             const float* __restrict__ cinv, int N)
{
    long long i = (long long)blockIdx.x * blockDim.x + threadIdx.x;
    if (i >= (long long)N * 128) return;
    int n = (int)(i >> 7), c = (int)(i & 127) * 4;
    size_t off = (size_t)n * 512 + c;
    float4 v1 = *(float4*)(x1 + off);
    float4 v2 = *(float4*)(x2 + off);
    float4 va = *(float4*)(agg + off);
    float4 bg = *(const float4*)(bgat + c);
    float4 bc = *(const float4*)(bgcn + c);
    float  ci = cinv[n];
    v1.x = elu1(v1.x + bg.x); v1.y = elu1(v1.y + bg.y);
    v1.z = elu1(v1.z + bg.z); v1.w = elu1(v1.w + bg.w);
    v2.x = elu1(v2.x + bc.x); v2.y = elu1(v2.y + bc.y);
    v2.z = elu1(v2.z + bc.z); v2.w = elu1(v2.w + bc.w);
    va.x *= ci; va.y *= ci; va.z *= ci; va.w *= ci;
    *(float4*)(x1 + off) = v1;
    *(float4*)(x2 + off) = v2;
    *(float4*)(agg + off) = va;
}

// y (in x1) = x1 + x2 + x3   (float4)
__global__ void ysum_kernel(float* x1, const float* __restrict__ x2,
                            const float* __restrict__ x3, int N)
{
    long long i = (long long)blockIdx.x * blockDim.x + threadIdx.x;
    if (i >= (long long)N * 128) return;
    size_t off = (size_t)i * 4;
    float4 a = *(float4*)(x1 + off);
    float4 b = *(const float4*)(x2 + off);
    float4 cc = *(const float4*)(x3 + off);
    a.x += b.x + cc.x; a.y += b.y + cc.y; a.z += b.z + cc.z; a.w += b.w + cc.w;
    *(float4*)(x1 + off) = a;
}

// out[n] = sum_k (gc[k]*RS*c[n,k] + bec[k]) * wc2[k] + bc2 ; one wave per node
__global__ __launch_bounds__(256) void final_kernel(
    const float* __restrict__ cbuf, const float* __restrict__ gc,
    const float* __restrict__ bec, const float* __restrict__ wc2,
    const float* __restrict__ bc2, float* __restrict__ out, int N)
{
    int wid  = (blockIdx.x * blockDim.x + threadIdx.x) >> 5;
    int lane = threadIdx.x & 31;
    if (wid >= N) return;
    const float* row = cbuf + (size_t)wid * 256;
    float s = 0.f;
#pragma unroll
    for (int k = lane; k < 256; k += 32)
        s += (gc[k] * BN_RS * row[k] + bec[k]) * wc2[k];
#pragma unroll
    for (int m = 16; m > 0; m >>= 1) s += __shfl_xor(s, m, 32);
    if (lane == 0) out[wid] = s + bc2[0];
}

// ---------------------------------------------------------------------------
// Host launcher
// ---------------------------------------------------------------------------
extern "C" void kernel_launch(void* const* d_in, const int* in_sizes, int n_in,
                              void* d_out, int out_size, void* d_ws, size_t ws_size,
                              hipStream_t stream)
{
    // Inputs in setup_inputs dict order (params flattened in insertion order)
    const float* x       = (const float*)d_in[0];
    const int*   ei      = (const int*)d_in[1];      // [2,E] row-major: src then dst
    const float* w_in    = (const float*)d_in[2];
    const float* b_in    = (const float*)d_in[3];
    const float* bn_in_g = (const float*)d_in[4];
    const float* bn_in_b = (const float*)d_in[5];
    const float* w_gat   = (const float*)d_in[6];
    const float* att_src = (const float*)d_in[7];
    const float* att_dst = (const float*)d_in[8];
    const float* b_gat   = (const float*)d_in[9];
    const float* w_gcn   = (const float*)d_in[10];
    const float* b_gcn   = (const float*)d_in[11];
    const float* w_sg_l  = (const float*)d_in[12];
    const float* w_sg_r  = (const float*)d_in[13];
    const float* b_sage  = (const float*)d_in[14];
    // res blocks: [w1,b1,g1,be1,w2,b2,g2,be2] x2 at 15..30
    const float* wc1     = (const float*)d_in[31];
    const float* bc1     = (const float*)d_in[32];
    const float* gc      = (const float*)d_in[33];
    const float* bec     = (const float*)d_in[34];
    const float* wc2     = (const float*)d_in[35];
    const float* bc2     = (const float*)d_in[36];

    const int D_IN = 1280, D_H = 512;
    const int N = in_sizes[0] / D_IN;
    const int E = in_sizes[1] / 2;
    float* out = (float*)d_out;

    // ---- workspace layout (floats) ----
    float* W = (float*)d_ws;
    size_t o = 0;
    float* h    = W + o; o += (size_t)N * D_H;
    float* hg   = W + o; o += (size_t)N * D_H;
    float* xw   = W + o; o += (size_t)N * D_H;
    float* tbuf = W + o; o += (size_t)N * D_H;   // sage tmp, then x3
    float* x1   = W + o; o += (size_t)N * D_H;   // ZERO; later y
    float* x2   = W + o; o += (size_t)N * D_H;   // ZERO; later t
    float* agg  = W + o; o += (size_t)N * D_H;   // ZERO; later cbuf (N*256 fits)
    float*    deg   = W + o; o += N;             // ZERO
    float*    cnt   = W + o; o += N;             // ZERO
    float*    denom = W + o; o += (size_t)N * 4; // ZERO
    unsigned* emax  = (unsigned*)(W + o); o += (size_t)N * 4; // ZERO (= mono -inf region)
    float* dinv  = W + o; o += N;
    float* cinv  = W + o; o += N;
    float* asrc  = W + o; o += (size_t)N * 4;
    float* adst  = W + o; o += (size_t)N * 4;
    float* ebuf  = W + o; o += (size_t)(E + N) * 4;   // logits -> ex -> alpha
    float* normE = W + o; o += (size_t)(E + N);
    (void)ws_size; (void)n_in; (void)out_size;

    // zero the accumulator region [x1 .. dinv) in one memset
    size_t zbytes = ((size_t)N * (3 * D_H + 1 + 1 + 4 + 4)) * sizeof(float);
    hipMemsetAsync((void*)x1, 0, zbytes, stream);

    auto cdiv = [](long long a, long long b) { return (unsigned)((a + b - 1) / b); };
    dim3 blk(256);

    // degrees + inverse norms
    deg_kernel<<<cdiv(E + N, 256), blk, 0, stream>>>(ei, deg, cnt, E, N);
    dinv_kernel<<<cdiv(N, 256), blk, 0, stream>>>(deg, cnt, dinv, cinv, N);

    // h = relu(bn(x @ w_in + b_in))
    gemm_wmma_f32<1, true, false, true><<<dim3(cdiv(N, 128), D_H / 64), blk, 0, stream>>>(
        x, w_in, b_in, bn_in_g, bn_in_b, nullptr, h, N, D_IN, D_H);

    // hg = h @ w_gat ; xw = h @ w_gcn
    gemm_wmma_f32<0, false, false, false><<<dim3(cdiv(N, 128), D_H / 64), blk, 0, stream>>>(
        h, w_gat, nullptr, nullptr, nullptr, nullptr, hg, N, D_H, D_H);
    gemm_wmma_f32<0, false, false, false><<<dim3(cdiv(N, 128), D_H / 64), blk, 0, stream>>>(
        h, w_gcn, nullptr, nullptr, nullptr, nullptr, xw, N, D_H, D_H);

    // attention scores + edge softmax
    att_kernel<<<cdiv((long long)N * 32, 256), blk, 0, stream>>>(hg, att_src, att_dst, asrc, adst, N);
    gat_logits_max<<<cdiv((long long)(E + N) * 4, 256), blk, 0, stream>>>(ei, asrc, adst, ebuf, emax, E, N);
    gat_expsum<<<cdiv((long long)(E + N) * 4, 256), blk, 0, stream>>>(ei, ebuf, emax, denom, E, N);
    gat_alpha<<<cdiv((long long)(E + N) * 4, 256), blk, 0, stream>>>(ei, ebuf, denom, dinv, normE, E, N);

    // fused GAT/GCN/SAGE aggregation (float4 channels)
    scatter_kernel<<<cdiv((long long)(E + N) * 128, 256), blk, 0, stream>>>(
        ei, hg, xw, h, ebuf, normE, x1, x2, agg, E, N);
    post_scatter<<<cdiv((long long)N * 128, 256), blk, 0, stream>>>(x1, x2, agg, b_gat, b_gcn, cinv, N);

    // SAGE: tbuf = agg @ w_sage_l ; x3 (in tbuf) = elu(h @ w_sage_r + tbuf + b_sage)
    gemm_wmma_f32<0, false, false, false><<<dim3(cdiv(N, 128), D_H / 64), blk, 0, stream>>>(
        agg, w_sg_l, nullptr, nullptr, nullptr, nullptr, tbuf, N, D_H, D_H);
    gemm_wmma_f32<2, false, true, true><<<dim3(cdiv(N, 128), D_H / 64), blk, 0, stream>>>(
        h, w_sg_r, b_sage, nullptr, nullptr, tbuf, tbuf, N, D_H, D_H);

    // y = x1 + x2 + x3   (y lives in x1)
    ysum_kernel<<<cdiv((long long)N * 128, 256), blk, 0, stream>>>(x1, x2, tbuf, N);

    // residual blocks: t = relu(bn(y@w1+b1)); y = relu(bn(t@w2+b2) + y)
    for (int rb = 0; rb < 2; ++rb) {
        const float* w1  = (const float*)d_in[15 + 8 * rb + 0];
        const float* b1  = (const float*)d_in[15 + 8 * rb + 1];
        const float* g1  = (const float*)d_in[15 + 8 * rb + 2];
        const float* be1 = (const float*)d_in[15 + 8 * rb + 3];
        const float* w2  = (const float*)d_in[15 + 8 * rb + 4];
        const float* b2  = (const float*)d_in[15 + 8 * rb + 5];
        const float* g2  = (const float*)d_in[15 + 8 * rb + 6];
        const float* be2 = (const float*)d_in[15 + 8 * rb + 7];
        gemm_wmma_f32<1, true, false, true><<<dim3(cdiv(N, 128), D_H / 64), blk, 0, stream>>>(
            x1, w1, b1, g1, be1, nullptr, x2, N, D_H, D_H);
        gemm_wmma_f32<1, true, true, true><<<dim3(cdiv(N, 128), D_H / 64), blk, 0, stream>>>(
            x2, w2, b2, g2, be2, x1, x1, N, D_H, D_H);
    }

    // classifier: cbuf (in agg) = relu(y @ wc1 + bc1) ; out = bn(cbuf) @ wc2 + bc2
    gemm_wmma_f32<1, false, false, true><<<dim3(cdiv(N, 128), 256 / 64), blk, 0, stream>>>(
        x1, wc1, bc1, nullptr, nullptr, nullptr, agg, N, D_H, 256);
    final_kernel<<<cdiv((long long)N * 32, 256), blk, 0, stream>>>(agg, gc, bec, wc2, bc2, out, N);
}